// CUDARenderModule_83906481095296
// MI455X (gfx1250) — compile-verified
//
#include <hip/hip_runtime.h>
#include <math.h>

// ---------------------------------------------------------------------------
// Gaussian-splat spherical volume render for MI455X (gfx1250, wave32).
// quad[t,r,g] = A t^2 + B t + C via V_WMMA_F32_16X16X4_F32 tiles
// (16 rays x 16 gaussians); exp + reduction on the VALU (co-executes with
// WMMA). Gaussian operands are double-buffer staged into LDS with
// global_load_async_to_lds_b128 (ASYNCcnt); WMMA operands are stored
// pair-interleaved so each operand is a single aligned ds_load_b64 /
// global_load_b64 straight into an even VGPR pair (no register shuffles).
// ---------------------------------------------------------------------------

typedef __attribute__((ext_vector_type(2))) float v2f;
typedef __attribute__((ext_vector_type(8))) float v8f;

#define G_N   2048
#define R_N   1024
#define T_N   64
#define GCH   128              // gaussians staged per LDS chunk
#define PI_F  3.14159265358979323846f
#define SH_C0 0.28209479177387814f
#define STEP  0.03125f

// chunk layout in LDS (floats): 6 pair-planes of 2*GCH, then C/opac/albedo
#define L_PAIR(p)  ((p) * 2 * GCH)
#define L_SCAL(s)  (6 * 2 * GCH + (s) * GCH)
#define L_TOT      (6 * 2 * GCH + 3 * GCH)     // 1920 floats
#define QTOT       (L_TOT / 4)                 // 480 b128 quads

// reduce across the 16 lanes of each wave32 half (xor masks stay in-half)
__device__ __forceinline__ float wsum16(float v) {
  v += __shfl_xor(v, 1, 32);
  v += __shfl_xor(v, 2, 32);
  v += __shfl_xor(v, 4, 32);
  v += __shfl_xor(v, 8, 32);
  return v;
}

// async copy 16B global -> LDS, tracked by ASYNCcnt (gfx1250)
__device__ __forceinline__ void async_copy_b128(const float* gsrc, float* lds_dst) {
  unsigned int       lds  = (unsigned int)(uintptr_t)lds_dst;  // low 32b of flat = LDS offset
  unsigned long long gadr = (unsigned long long)(uintptr_t)gsrc;
  asm volatile("global_load_async_to_lds_b128 %0, %1, off"
               :: "v"(lds), "v"(gadr) : "memory");
}

__device__ __forceinline__ void wait_async0() {
  asm volatile("s_wait_asynccnt 0" ::: "memory");
}

// ---------------------------------------------------------------------------
// Kernel 0: per-gaussian preprocessing. Feature rows (K-dim):
// 0:Pxx 1:Pyy 2:Pzz 3:2Pxy 4:2Pxz 5:2Pyz 6:0 7:0 8..10:2(P om) 11:0,
// stored pair-interleaved: gFi[p][g][2] holds rows (2p, 2p+1).
// ---------------------------------------------------------------------------
__global__ __launch_bounds__(256) void k_gauss_pre(
    const float* __restrict__ means, const float* __restrict__ scales,
    const float* __restrict__ rots,  const float* __restrict__ opacs,
    const float* __restrict__ feats, const float* __restrict__ cam,
    float* __restrict__ gFi, float* __restrict__ gC,
    float* __restrict__ gO, float* __restrict__ gA) {
  int g = blockIdx.x * blockDim.x + threadIdx.x;
  if (g >= G_N) return;

  float a0 = __expf(scales[3*g+0]);
  float a1 = __expf(scales[3*g+1]);
  float a2 = __expf(scales[3*g+2]);
  a0 = 1.0f / (a0*a0); a1 = 1.0f / (a1*a1); a2 = 1.0f / (a2*a2);

  float qw = rots[4*g+0], qx = rots[4*g+1], qy = rots[4*g+2], qz = rots[4*g+3];
  float qn = sqrtf(qw*qw + qx*qx + qy*qy + qz*qz) + 1e-12f;
  qw /= qn; qx /= qn; qy /= qn; qz /= qn;

  float r00 = 1.f - 2.f*(qy*qy + qz*qz), r01 = 2.f*(qx*qy - qw*qz), r02 = 2.f*(qx*qz + qw*qy);
  float r10 = 2.f*(qx*qy + qw*qz), r11 = 1.f - 2.f*(qx*qx + qz*qz), r12 = 2.f*(qy*qz - qw*qx);
  float r20 = 2.f*(qx*qz - qw*qy), r21 = 2.f*(qy*qz + qw*qx), r22 = 1.f - 2.f*(qx*qx + qy*qy);

  // P = R diag(a) R^T (symmetric)
  float p00 = r00*r00*a0 + r01*r01*a1 + r02*r02*a2;
  float p01 = r00*r10*a0 + r01*r11*a1 + r02*r12*a2;
  float p02 = r00*r20*a0 + r01*r21*a1 + r02*r22*a2;
  float p11 = r10*r10*a0 + r11*r11*a1 + r12*r12*a2;
  float p12 = r10*r20*a0 + r11*r21*a1 + r12*r22*a2;
  float p22 = r20*r20*a0 + r21*r21*a1 + r22*r22*a2;

  float om0 = cam[0] - means[3*g+0];
  float om1 = cam[1] - means[3*g+1];
  float om2 = cam[2] - means[3*g+2];

  float pm0 = p00*om0 + p01*om1 + p02*om2;
  float pm1 = p01*om0 + p11*om1 + p12*om2;
  float pm2 = p02*om0 + p12*om1 + p22*om2;

  v2f pv;
  pv.x = p00;      pv.y = p11;      *(v2f*)&gFi[(0*G_N + g)*2] = pv;
  pv.x = p22;      pv.y = 2.f*p01;  *(v2f*)&gFi[(1*G_N + g)*2] = pv;
  pv.x = 2.f*p02;  pv.y = 2.f*p12;  *(v2f*)&gFi[(2*G_N + g)*2] = pv;
  pv.x = 0.f;      pv.y = 0.f;      *(v2f*)&gFi[(3*G_N + g)*2] = pv;
  pv.x = 2.f*pm0;  pv.y = 2.f*pm1;  *(v2f*)&gFi[(4*G_N + g)*2] = pv;
  pv.x = 2.f*pm2;  pv.y = 0.f;      *(v2f*)&gFi[(5*G_N + g)*2] = pv;

  gC[g] = om0*pm0 + om1*pm1 + om2*pm2;
  gO[g] = 1.0f / (1.0f + __expf(-opacs[g]));

  float f0 = fmaxf(SH_C0*feats[3*g+0] + 0.5f, 0.f);
  float f1 = fmaxf(SH_C0*feats[3*g+1] + 0.5f, 0.f);
  float f2 = fmaxf(SH_C0*feats[3*g+2] + 0.5f, 0.f);
  gA[g] = (f0 + f1 + f2) * (1.0f/3.0f);
}

// ---------------------------------------------------------------------------
// Kernel 1: per-ray features, pair-interleaved rayFi[p][r][2], rows:
// 0:dx2 1:dy2 2:dz2 3:dxdy 4:dxdz 5:dydz 6:0 7:0 8:dx 9:dy 10:dz 11:0
// ---------------------------------------------------------------------------
__global__ __launch_bounds__(256) void k_ray_pre(float* __restrict__ rayFi) {
  int r = blockIdx.x * blockDim.x + threadIdx.x;
  if (r >= R_N) return;
  int it = r >> 5, ip = r & 31;
  float th = (float)it * ((PI_F * 0.5f) / 31.0f);
  float ph = -PI_F * 0.5f + (float)ip * (PI_F / 31.0f);
  float st = sinf(th), ct = cosf(th);
  float dx = st * cosf(ph), dy = st * sinf(ph), dz = ct;

  v2f pv;
  pv.x = dx*dx;  pv.y = dy*dy;  *(v2f*)&rayFi[(0*R_N + r)*2] = pv;
  pv.x = dz*dz;  pv.y = dx*dy;  *(v2f*)&rayFi[(1*R_N + r)*2] = pv;
  pv.x = dx*dz;  pv.y = dy*dz;  *(v2f*)&rayFi[(2*R_N + r)*2] = pv;
  pv.x = 0.f;    pv.y = 0.f;    *(v2f*)&rayFi[(3*R_N + r)*2] = pv;
  pv.x = dx;     pv.y = dy;     *(v2f*)&rayFi[(4*R_N + r)*2] = pv;
  pv.x = dz;     pv.y = 0.f;    *(v2f*)&rayFi[(5*R_N + r)*2] = pv;
}

// ---------------------------------------------------------------------------
// Kernel 2: main loop. One wave32 per (t, 16-ray tile): 64*64 = 4096 waves,
// 8 waves/block; all waves in a block share t and walk the same gaussians.
// 128-gaussian chunks double-buffer staged into LDS with async b128 copies;
// per 16-gaussian sub-tile: 3x v_wmma_f32_16x16x4_f32 with operands fetched
// as single ds_load_b64 each (pair-interleaved layout), then per-lane
// exp/accumulate. C/D layout: lane l -> N=l&15 (gaussian),
// VGPR i -> M = i + 8*(l>>4) (ray). Final reduce over N via shfl_xor.
// K-block kb uses K rows {2h, 2h+1} (h = lane>>4) -> pair index (kb>>1)+h.
// ---------------------------------------------------------------------------
__global__ __launch_bounds__(256) void k_main(
    const float* __restrict__ rayFi, const float* __restrict__ gFi,
    const float* __restrict__ gC,    const float* __restrict__ gO,
    const float* __restrict__ gA,
    float* __restrict__ dens, float* __restrict__ rho) {
  __shared__ __align__(16) float sbuf[2][L_TOT];

  int tid  = threadIdx.x;
  int wid  = blockIdx.x * 8 + (tid >> 5);             // 0..4095
  int lane = tid & 31;
  int half = lane >> 4;
  int col  = lane & 15;
  int tIdx  = wid >> 6;          // 0..63 (uniform within block)
  int rbase = (wid & 63) << 4;   // 16-ray tile base

  float tval = 0.5f + (float)tIdx * (2.0f / 63.0f);
  float t2 = tval * tval;

  // A operand: one aligned b64 per K-block (pair index (kb>>1)+half)
  int arow = rbase + col;
  v2f a0 = *(const v2f*)&rayFi[((0 + half)*R_N + arow)*2];
  v2f a1 = *(const v2f*)&rayFi[((2 + half)*R_N + arow)*2];
  v2f a2 = *(const v2f*)&rayFi[((4 + half)*R_N + arow)*2];

  float dacc[8], racc[8];
#pragma unroll
  for (int i = 0; i < 8; ++i) { dacc[i] = 0.f; racc[i] = 0.f; }

  // cooperative async stage of one 128-gaussian chunk into sbuf[b]:
  // quads 0..383 cover the 6 pair-planes, 384..479 the 3 scalar planes
  auto issue_stage = [&](int gb, int b) {
#pragma unroll
    for (int j0 = 0; j0 < 2; ++j0) {
      int j = tid + j0 * 256;
      if (j < QTOT) {
        const float* src;
        int ldsoff;
        if (j < 384) {
          int ps = j >> 6;                 // pair plane 0..5
          int qf = (j & 63) << 2;          // float offset within plane chunk
          src    = gFi + (ps * G_N + gb) * 2 + qf;
          ldsoff = L_PAIR(ps) + qf;
        } else {
          int jj = j - 384;
          int s  = jj >> 5;                // 0:C 1:opac 2:albedo
          int qf = (jj & 31) << 2;
          src    = (s == 0 ? gC : (s == 1 ? gO : gA)) + gb + qf;
          ldsoff = L_SCAL(s) + qf;
        }
        async_copy_b128(src, &sbuf[b][ldsoff]);
      }
    }
  };

  issue_stage(0, 0);
  wait_async0();
  __syncthreads();

  for (int s = 0; s < G_N / GCH; ++s) {
    if (s + 1 < G_N / GCH) issue_stage((s + 1) * GCH, (s + 1) & 1);

    const float* sb = sbuf[s & 1];
#pragma unroll
    for (int sub = 0; sub < GCH / 16; ++sub) {
      int c = sub * 16 + col;
      v2f b0 = *(const v2f*)&sb[L_PAIR(0 + half) + c * 2];
      v2f b1 = *(const v2f*)&sb[L_PAIR(2 + half) + c * 2];
      v2f b2 = *(const v2f*)&sb[L_PAIR(4 + half) + c * 2];
      float Cg = sb[L_SCAL(0) + c];
      float og = sb[L_SCAL(1) + c];
      float ag = sb[L_SCAL(2) + c];

      v8f accA = {};
      accA = __builtin_amdgcn_wmma_f32_16x16x4_f32(
          false, a0, false, b0, (short)0, accA, false, false);
      accA = __builtin_amdgcn_wmma_f32_16x16x4_f32(
          false, a1, false, b1, (short)0, accA, false, false);
      v8f accB = {};
      accB = __builtin_amdgcn_wmma_f32_16x16x4_f32(
          false, a2, false, b2, (short)0, accB, false, false);

#pragma unroll
      for (int i = 0; i < 8; ++i) {
        float quad = accA[i] * t2 + accB[i] * tval + Cg;
        float w = og * __expf(-0.5f * quad);
        dacc[i] += w;
        racc[i] += w * ag;
      }
    }

    wait_async0();      // our wave's async loads for next chunk have landed
    __syncthreads();    // everyone's have; safe to swap buffers
  }

#pragma unroll
  for (int i = 0; i < 8; ++i) { dacc[i] = wsum16(dacc[i]); racc[i] = wsum16(racc[i]); }

  if (col == 0) {
#pragma unroll
    for (int i = 0; i < 8; ++i) {
      int row = rbase + i + 8 * half;
      dens[tIdx * R_N + row] = dacc[i];
      rho [tIdx * R_N + row] = racc[i];
    }
  }
}

// ---------------------------------------------------------------------------
// Kernel 3: per-ray exclusive cumsum -> transmittance -> result
// ---------------------------------------------------------------------------
__global__ __launch_bounds__(256) void k_finalize(
    const float* __restrict__ dens, const float* __restrict__ rho,
    float* __restrict__ out) {
  int r = blockIdx.x * blockDim.x + threadIdx.x;
  if (r >= R_N) return;
  int it = r >> 5;
  float th = (float)it * ((PI_F * 0.5f) / 31.0f);
  float sth = sinf(th);
  float acc = 0.f;
  for (int k = 0; k < T_N; ++k) {
    float t = 0.5f + (float)k * (2.0f / 63.0f);
    float trans = __expf(-acc);
    float rd = rho[k * R_N + r] * trans;
    out[k * R_N + r] = rd / (t * t + 1e-8f) * sth;
    acc += dens[k * R_N + r] * STEP;  // exclusive cumsum
  }
}

// ---------------------------------------------------------------------------
// Kernel 4: histogram over (theta,phi) per radial slice (LDS tree reduce)
// ---------------------------------------------------------------------------
__global__ __launch_bounds__(256) void k_hist(
    const float* __restrict__ result, float* __restrict__ hist) {
  __shared__ float sm[256];
  int k = blockIdx.x;
  float s = 0.f;
  for (int j = threadIdx.x; j < R_N; j += 256) s += result[k * R_N + j];
  sm[threadIdx.x] = s;
  __syncthreads();
  for (int off = 128; off > 0; off >>= 1) {
    if ((int)threadIdx.x < off) sm[threadIdx.x] += sm[threadIdx.x + off];
    __syncthreads();
  }
  if (threadIdx.x == 0) {
    float dth = (PI_F * 0.5f) / 32.0f;
    float dph = PI_F / 32.0f;
    hist[k] = sm[0] * dth * dph;
  }
}

// ---------------------------------------------------------------------------
extern "C" void kernel_launch(void* const* d_in, const int* in_sizes, int n_in,
                              void* d_out, int out_size, void* d_ws, size_t ws_size,
                              hipStream_t stream) {
  const float* means  = (const float*)d_in[0];
  const float* scales = (const float*)d_in[1];
  const float* rots   = (const float*)d_in[2];
  const float* opacs  = (const float*)d_in[3];
  const float* feats  = (const float*)d_in[4];
  const float* cam    = (const float*)d_in[5];
  (void)in_sizes; (void)n_in; (void)out_size; (void)ws_size;

  float* ws   = (float*)d_ws;
  float* gFi  = ws;                    // 6 * 2048 * 2 = 12 * G_N
  float* gC   = gFi  + 12 * G_N;       // 2048
  float* gO   = gC   + G_N;            // 2048
  float* gA   = gO   + G_N;            // 2048
  float* rayF = gA   + G_N;            // 6 * 1024 * 2 = 12 * R_N
  float* dens = rayF + 12 * R_N;       // 64 * 1024
  float* rho  = dens + T_N * R_N;      // 64 * 1024

  float* out  = (float*)d_out;         // result: 65536 floats, hist: 64 floats

  k_gauss_pre<<<dim3(G_N / 256), dim3(256), 0, stream>>>(
      means, scales, rots, opacs, feats, cam, gFi, gC, gO, gA);
  k_ray_pre<<<dim3(R_N / 256), dim3(256), 0, stream>>>(rayF);
  k_main<<<dim3(512), dim3(256), 0, stream>>>(rayF, gFi, gC, gO, gA, dens, rho);
  k_finalize<<<dim3(R_N / 256), dim3(256), 0, stream>>>(dens, rho, out);
  k_hist<<<dim3(T_N), dim3(256), 0, stream>>>(out, out + T_N * R_N);
}